// RecurrentChannelAttentionV2_50998441672741
// MI455X (gfx1250) — compile-verified
//
#include <hip/hip_runtime.h>
#include <math.h>

// Problem dims (fixed by the reference)
#define CC 512
#define TT 16384
#define MM 5
#define OMF  0.95f      // 1 - momentum
#define MOMF 0.05f
#define EPSF 1e-12f

// LDS operand ring: NSLOT slots, prefetch PF steps ahead (PF < NSLOT so the
// slot being overwritten was consumed >=1 step ago, past a barrier).
#define NSLOT 5
#define PF 4
#define SLOT_F (CC*MM + CC*MM + CC)   // K slice + V slice + D slice = 5632 floats
#define RED_F 32                       // 2 x 16 wave partials (ping-pong)

typedef unsigned int       u32;
typedef unsigned long long u64;
typedef unsigned int u32x4 __attribute__((ext_vector_type(4)));
typedef int          i32x4 __attribute__((ext_vector_type(4)));
typedef int          i32x8 __attribute__((ext_vector_type(8)));

typedef __attribute__((address_space(1))) int glb_i32;
typedef __attribute__((address_space(3))) int lds_i32;

#if defined(__has_builtin)
#if __has_builtin(__builtin_amdgcn_tensor_load_to_lds) && __has_builtin(__builtin_amdgcn_s_wait_tensorcnt)
#define USE_TDM 1
#endif
#endif
#ifndef USE_TDM
#define USE_TDM 0
#endif

#if USE_TDM
// ---- Tensor Data Mover path: one DMA descriptor per (K|V|D) step slice ----
// D# group0 (ISA 08 §8.3): [1:0]=count=1, [63:32]=lds_addr, [120:64]=global_addr,
// [127:126]=type=2.  group1 (§8.4) packs dims/strides; groups 2/3 zero (2D tile).
__device__ __forceinline__ void tdm_load_2d(u64 gaddr, u32 lds_addr, i32x8 g1) {
  u32x4 g0;
  g0[0] = 1u;                                            // count=1, user mode
  g0[1] = lds_addr;                                      // LDS byte address
  g0[2] = (u32)gaddr;                                    // global_addr[31:0]
  g0[3] = (u32)((gaddr >> 32) & 0x01FFFFFFull) | (2u << 30);  // addr[56:32] | type=2
  i32x4 z4 = {0, 0, 0, 0};
  i32x8 z8 = {0, 0, 0, 0, 0, 0, 0, 0};
  // clang-23 / therock arity: (g0, g1, g2, g3, extra, cpol)
  __builtin_amdgcn_tensor_load_to_lds(g0, g1, z4, z4, z8, 0);
}
// group1 for a (rows=512) x (row_len) f32 tile with row stride `stride` elems:
//   dw0: data_size=2 (4B) at bits[17:16]            -> 0x00020000
//   dw1[31:16]=tensor_dim0[15:0], dw2[31:16]=tensor_dim1[15:0]=512
//   dw3[31:16]=tile_dim0, dw4[15:0]=tile_dim1=512, dw5=dim0_stride[31:0]
__device__ __forceinline__ i32x8 make_g1(int row_len, int stride) {
  i32x8 g1;
  g1[0] = 0x00020000;
  g1[1] = row_len << 16;
  g1[2] = (int)(512u << 16);
  g1[3] = row_len << 16;
  g1[4] = 512;
  g1[5] = stride;
  g1[6] = 0;
  g1[7] = 0;
  return g1;
}
__device__ __forceinline__ void issue_step_tdm(u64 Kb, u64 Vb, u64 Db, u32 slot_lds,
                                               int t, i32x8 g1kv, i32x8 g1d) {
  tdm_load_2d(Kb + (u64)t * (MM * 4), slot_lds, g1kv);
  tdm_load_2d(Vb + (u64)t * (MM * 4), slot_lds + CC * MM * 4, g1kv);
  tdm_load_2d(Db + (u64)t * 4, slot_lds + 2 * CC * MM * 4, g1d);
}
#else
// ---- Fallback: per-lane ASYNCcnt-tracked async loads ----
__device__ __forceinline__ void async_ld_b32(const float* g, float* l) {
#if defined(__has_builtin) && __has_builtin(__builtin_amdgcn_global_load_async_to_lds_b32)
  __builtin_amdgcn_global_load_async_to_lds_b32((glb_i32*)g, (lds_i32*)l, 0, 0);
#else
  unsigned loff = (unsigned)(unsigned long long)(__attribute__((address_space(3))) float*)l;
  unsigned long long ga = (unsigned long long)g;
  asm volatile("global_load_async_to_lds_b32 %0, %1, off"
               :: "v"(loff), "v"(ga) : "memory");
#endif
}
__device__ __forceinline__ void issue_step_async(const float* Kc, const float* Vc,
                                                 const float* Dc, float* slot,
                                                 int t, int c) {
  const float* kp = Kc + (size_t)t * MM;
  const float* vp = Vc + (size_t)t * MM;
  float* lk = slot + c * MM;
  float* lv = slot + CC * MM + c * MM;
#pragma unroll
  for (int j = 0; j < MM; ++j) async_ld_b32(kp + j, lk + j);
#pragma unroll
  for (int j = 0; j < MM; ++j) async_ld_b32(vp + j, lv + j);
  async_ld_b32(Dc + t, slot + 2 * CC * MM + c);
}
__device__ __forceinline__ void wait_async_le33() {
#if defined(__has_builtin) && __has_builtin(__builtin_amdgcn_s_wait_asynccnt)
  __builtin_amdgcn_s_wait_asynccnt(33);
#else
  asm volatile("s_wait_asynccnt 33" ::: "memory");
#endif
  asm volatile("" ::: "memory");
}
#endif

__global__ __launch_bounds__(512, 1)
void ca_scan_kernel(const float* __restrict__ K, const float* __restrict__ V,
                    const float* __restrict__ D, float* __restrict__ out) {
  __shared__ float smem[NSLOT * SLOT_F + RED_F];   // 112,768 B of the 320 KB WGP LDS
  const int c    = threadIdx.x;          // one thread per channel (wave32 x 16 waves)
  const int wave = c >> 5;
  const int lane = c & 31;
  float* red = smem + NSLOT * SLOT_F;

  const float* Kc = K + (size_t)c * (TT * MM);
  const float* Vc = V + (size_t)c * (TT * MM);
  const float* Dc = D + (size_t)c * TT;
  float*     outc = out + (size_t)c * TT;     // out layout (MQ=1, C, T)

#if USE_TDM
  const u32 smem_lds = (u32)(size_t)(__attribute__((address_space(3))) float*)smem;
  const u64 Kb = (u64)(size_t)K;
  const u64 Vb = (u64)(size_t)V;
  const u64 Db = (u64)(size_t)D;
  const i32x8 g1kv = make_g1(MM, TT * MM);   // 512 x 5  f32, row stride 81920
  const i32x8 g1d  = make_g1(1, TT);         // 512 x 1  f32, row stride 16384
  // Prime the DMA pipeline: wave 0 issues steps 0..PF-1 (3 descriptors each).
  if (c < 32) {
    for (int p = 0; p < PF; ++p)
      issue_step_tdm(Kb, Vb, Db, smem_lds + (u32)(p * SLOT_F * 4), p, g1kv, g1d);
    __builtin_amdgcn_s_wait_tensorcnt(3 * (PF - 1));   // slot 0 complete
  }
  __syncthreads();
#else
  for (int p = 0; p < PF; ++p)
    issue_step_async(Kc, Vc, Dc, smem + p * SLOT_F, p, c);
#endif

  // EMA raw state (5 regs/thread) + carried normalization scale.
  float s0 = 0.f, s1 = 0.f, s2 = 0.f, s3 = 0.f, s4 = 0.f;
  float scale = 0.f;       // y(t) uses att from step t-1; att(-1)=0 -> s==0 anyway
  float pw = OMF;          // om^n with n starting at 1

  int slot_idx = 0;
  for (int t = 0; t < TT; ++t) {
#if !USE_TDM
    wait_async_le33();     // oldest ring slot's 11 async loads are complete
#endif
    float* slot = smem + slot_idx * SLOT_F;

    const float k0 = slot[c * MM + 0], k1 = slot[c * MM + 1], k2 = slot[c * MM + 2],
                k3 = slot[c * MM + 3], k4 = slot[c * MM + 4];
    const float v0 = slot[CC * MM + c * MM + 0], v1 = slot[CC * MM + c * MM + 1],
                v2 = slot[CC * MM + c * MM + 2], v3 = slot[CC * MM + c * MM + 3],
                v4 = slot[CC * MM + c * MM + 4];
    const float d  = slot[2 * CC * MM + c];

    // y = att_prev . v  where att_prev = s / (sqrt(ss) + bias*eps), folded into scale
    const float y = scale * (s0 * v0 + s1 * v1 + s2 * v2 + s3 * v3 + s4 * v4);
    outc[t] = y;

    // EMA rank-1 update
    const float e = d + y;
    const float me = MOMF * e;
    s0 = OMF * s0 + me * k0;
    s1 = OMF * s1 + me * k1;
    s2 = OMF * s2 + me * k2;
    s3 = OMF * s3 + me * k3;
    s4 = OMF * s4 + me * k4;

    // Global sum of squares over all C*M state entries (couples all channels)
    float ssc = s0 * s0 + s1 * s1 + s2 * s2 + s3 * s3 + s4 * s4;
    ssc += __shfl_xor(ssc, 16, 32);
    ssc += __shfl_xor(ssc,  8, 32);
    ssc += __shfl_xor(ssc,  4, 32);
    ssc += __shfl_xor(ssc,  2, 32);
    ssc += __shfl_xor(ssc,  1, 32);
    float* rp = red + (t & 1) * 16;      // ping-pong so one barrier/step suffices
    if (lane == 0) rp[wave] = ssc;

    // Kick the DMA for step t+PF; slot (t+PF)%NSLOT was consumed at step t-1,
    // past a barrier, so overwrite is safe.
    const int tp = t + PF;
    int ns = slot_idx + PF; if (ns >= NSLOT) ns -= NSLOT;
#if USE_TDM
    if (c < 32) {                       // wave-uniform: wave 0 drives the TDM
      if (tp < TT)
        issue_step_tdm(Kb, Vb, Db, smem_lds + (u32)(ns * SLOT_F * 4), tp, g1kv, g1d);
      // <=9 outstanding => slot t+1's three loads retired (TDM in-order per wave)
      __builtin_amdgcn_s_wait_tensorcnt(3 * (PF - 1));
    }
#else
    if (tp < TT) issue_step_async(Kc, Vc, Dc, smem + ns * SLOT_F, tp, c);
#endif

    __syncthreads();

    float ss = 0.f;
#pragma unroll
    for (int w = 0; w < 16; ++w) ss += rp[w];

    pw *= OMF;                             // om^(t+2)
    const float bias = 1.f - pw;           // unbias factor
    scale = 1.f / (sqrtf(ss) + bias * EPSF);

    slot_idx = (slot_idx + 1 == NSLOT) ? 0 : slot_idx + 1;
  }
}

extern "C" void kernel_launch(void* const* d_in, const int* in_sizes, int n_in,
                              void* d_out, int out_size, void* d_ws, size_t ws_size,
                              hipStream_t stream) {
  const float* K = (const float*)d_in[0];   // (C,T,M) f32
  const float* V = (const float*)d_in[1];   // (C,T,M) f32
  const float* D = (const float*)d_in[2];   // (C,T,MQ=1) f32
  float* out = (float*)d_out;               // (1,C,T) f32

  // Single workgroup owns the whole recurrence (global per-step norm).
  ca_scan_kernel<<<1, CC, 0, stream>>>(K, V, D, out);
}